// OhemPairTripletLoss_49546742727249
// MI455X (gfx1250) — compile-verified
//
#include <hip/hip_runtime.h>

typedef __attribute__((ext_vector_type(2))) float v2f;
typedef __attribute__((ext_vector_type(8))) float v8f;

#define NROWS   8192
#define DIMS    128
#define MARGINF 0.3f
#define TOPK    512u   // top-1024 of doubled set == top-512 distinct, each twice

// ---- workspace layout (32-bit words) ----
#define H_SUM   0   // float: sum of off-diagonal relu losses
#define H_MAX   1   // uint : max loss as ordered float bits
#define H_CNT   2   // uint : candidate count
#define H_THR   3   // uint : collect threshold (float bits, bucket lower bound)
#define H_K     4   // uint : k = min(512, #positives)
#define H_P     5   // uint : total positive count
#define OFF_HIST 16
#define NBINS    8192
#define OFF_SQ1  (OFF_HIST + NBINS)
#define OFF_SQ2  (OFF_SQ1 + NROWS)
#define OFF_APM  (OFF_SQ2 + NROWS)
#define OFF_CAND (OFF_APM + NROWS)

__device__ __forceinline__ v8f wmma_f32_k4(v2f a, v2f b, v8f c) {
    // D = A(16x4 f32) * B(4x16 f32) + C(16x16 f32)
    return __builtin_amdgcn_wmma_f32_16x16x4_f32(false, a, false, b, (short)0, c,
                                                 false, false);
}

// -------------------- init: zero header + global histogram --------------------
__global__ void init_kernel(unsigned int* __restrict__ ws) {
    int i = blockIdx.x * blockDim.x + threadIdx.x;
    int total = OFF_HIST + NBINS;
    for (; i < total; i += gridDim.x * blockDim.x) ws[i] = 0u;
}

// -------------------- prep: row norms + (dist_ap + margin) --------------------
// one wave32 per row; each lane reads a float4 (32*4 = 128 = DIMS)
__global__ __launch_bounds__(256) void prep_kernel(const float* __restrict__ p1,
                                                   const float* __restrict__ p2,
                                                   unsigned int* __restrict__ ws) {
    const int lane = threadIdx.x & 31;
    const int wid  = threadIdx.x >> 5;
    const int row  = blockIdx.x * 8 + wid;
    if (row >= NROWS) return;

    const float4 a = *(const float4*)(p1 + (size_t)row * DIMS + lane * 4);
    const float4 b = *(const float4*)(p2 + (size_t)row * DIMS + lane * 4);
    float s1 = a.x*a.x + a.y*a.y + a.z*a.z + a.w*a.w;
    float s2 = b.x*b.x + b.y*b.y + b.z*b.z + b.w*b.w;
    float dx = a.x-b.x, dy = a.y-b.y, dz = a.z-b.z, dw = a.w-b.w;
    float sd = dx*dx + dy*dy + dz*dz + dw*dw;
    for (int o = 16; o; o >>= 1) {
        s1 += __shfl_xor(s1, o, 32);
        s2 += __shfl_xor(s2, o, 32);
        sd += __shfl_xor(sd, o, 32);
    }
    if (lane == 0) {
        ((float*)ws)[OFF_SQ1 + row] = s1;
        ((float*)ws)[OFF_SQ2 + row] = s2;
        ((float*)ws)[OFF_APM + row] = sd + MARGINF;  // dist_ap + margin
    }
}

// -------------------- fused GEMM + epilogue --------------------
// MODE 0: accumulate sum / max / histogram.  MODE 1: collect values >= threshold.
// Block = 256 thr = 8 waves; wave tile 32(M) x 64(N); block tile 128 x 128.
template <int MODE>
__global__ __launch_bounds__(256) void gemm_kernel(const float* __restrict__ p1,
                                                   const float* __restrict__ p2,
                                                   unsigned int* __restrict__ ws,
                                                   float* __restrict__ cand,
                                                   int candCap) {
    __shared__ unsigned int lhist[(MODE == 0) ? NBINS : 1];
    __shared__ float wsum[8];
    __shared__ unsigned int wmax[8];

    const int tid  = threadIdx.x;
    const int lane = tid & 31;
    const int wid  = tid >> 5;
    const int wy   = wid >> 1;       // 0..3 -> M sub-tile (32 rows)
    const int wx   = wid & 1;        // 0..1 -> N sub-tile (64 cols)
    const int bm   = blockIdx.y * 128;
    const int bn   = blockIdx.x * 128;

    if (MODE == 0) {
        for (int i = tid; i < NBINS; i += 256) lhist[i] = 0u;
        __syncthreads();
    }

    const int laneM = lane & 15;           // M (for A) / N (for B) within 16
    const int kq    = (lane >> 4) * 2;     // lanes 16-31 carry K = 2,3
    const int rowA  = bm + wy * 32 + laneM;

    const float* __restrict__ pA0 = p1 + (size_t)rowA * DIMS + kq;
    const float* __restrict__ pA1 = pA0 + (size_t)16 * DIMS;
    const float* __restrict__ pB0 = p2 + (size_t)(bn + wx * 64 + 0  + laneM) * DIMS + kq;
    const float* __restrict__ pB1 = p2 + (size_t)(bn + wx * 64 + 16 + laneM) * DIMS + kq;
    const float* __restrict__ pB2 = p2 + (size_t)(bn + wx * 64 + 32 + laneM) * DIMS + kq;
    const float* __restrict__ pB3 = p2 + (size_t)(bn + wx * 64 + 48 + laneM) * DIMS + kq;

    v8f acc[2][4] = {};
#pragma unroll 4
    for (int k0 = 0; k0 < DIMS; k0 += 4) {
        v2f a0 = *(const v2f*)(pA0 + k0);
        v2f a1 = *(const v2f*)(pA1 + k0);
        v2f b0 = *(const v2f*)(pB0 + k0);
        v2f b1 = *(const v2f*)(pB1 + k0);
        v2f b2 = *(const v2f*)(pB2 + k0);
        v2f b3 = *(const v2f*)(pB3 + k0);
        acc[0][0] = wmma_f32_k4(a0, b0, acc[0][0]);
        acc[0][1] = wmma_f32_k4(a0, b1, acc[0][1]);
        acc[0][2] = wmma_f32_k4(a0, b2, acc[0][2]);
        acc[0][3] = wmma_f32_k4(a0, b3, acc[0][3]);
        acc[1][0] = wmma_f32_k4(a1, b0, acc[1][0]);
        acc[1][1] = wmma_f32_k4(a1, b1, acc[1][1]);
        acc[1][2] = wmma_f32_k4(a1, b2, acc[1][2]);
        acc[1][3] = wmma_f32_k4(a1, b3, acc[1][3]);
    }

    // ---- epilogue ----
    const float* sq1 = (const float*)ws + OFF_SQ1;
    const float* sq2 = (const float*)ws + OFF_SQ2;
    const float* apm = (const float*)ws + OFF_APM;
    const int half = lane >> 4;   // C layout: VGPR r -> rows r (lanes 0-15), r+8 (16-31)
    const int nl   = lane & 15;

    float s1v[2][8], apv[2][8];
#pragma unroll
    for (int mi = 0; mi < 2; ++mi)
#pragma unroll
        for (int r = 0; r < 8; ++r) {
            int m = bm + wy * 32 + mi * 16 + half * 8 + r;
            s1v[mi][r] = sq1[m];
            apv[mi][r] = apm[m];
        }

    unsigned int thr = 0u;
    if (MODE == 1) thr = ws[H_THR];

    float lsum = 0.f;
    unsigned int lmax = 0u;
#pragma unroll
    for (int ni = 0; ni < 4; ++ni) {
        const int n = bn + wx * 64 + ni * 16 + nl;
        const float s2n = sq2[n];
#pragma unroll
        for (int mi = 0; mi < 2; ++mi) {
#pragma unroll
            for (int r = 0; r < 8; ++r) {
                const int m = bm + wy * 32 + mi * 16 + half * 8 + r;
                float d = s1v[mi][r] + s2n - 2.0f * acc[mi][ni][r];
                float lossv = fmaxf(apv[mi][r] - d, 0.0f);  // apm includes +margin
                if (m != n) {
                    if (MODE == 0) {
                        lsum += lossv;
                        unsigned int bits = __float_as_uint(lossv);
                        if (bits) {
                            lmax = max(lmax, bits);
                            atomicAdd(&lhist[bits >> 18], 1u);
                        }
                    } else {
                        unsigned int bits = __float_as_uint(lossv);
                        if (lossv > 0.0f && bits >= thr) {
                            unsigned int idx = atomicAdd(&ws[H_CNT], 1u);
                            if (idx < (unsigned int)candCap) cand[idx] = lossv;
                        }
                    }
                }
            }
        }
    }

    if (MODE == 0) {
        for (int o = 16; o; o >>= 1) {
            lsum += __shfl_xor(lsum, o, 32);
            lmax = max(lmax, (unsigned int)__shfl_xor((int)lmax, o, 32));
        }
        if (lane == 0) { wsum[wid] = lsum; wmax[wid] = lmax; }
        __syncthreads();
        if (tid == 0) {
            float bs = 0.f; unsigned int bm2 = 0u;
            for (int w = 0; w < 8; ++w) { bs += wsum[w]; bm2 = max(bm2, wmax[w]); }
            atomicAdd((float*)ws + H_SUM, bs);
            atomicMax(&ws[H_MAX], bm2);
        }
        for (int i = tid; i < NBINS; i += 256)
            if (lhist[i]) atomicAdd(&ws[OFF_HIST + i], lhist[i]);
    }
}

// -------------------- threshold from histogram --------------------
__global__ void threshold_kernel(unsigned int* __restrict__ ws) {
    if (threadIdx.x != 0) return;
    unsigned long long P = 0;
    for (int b = 1; b < NBINS; ++b) P += ws[OFF_HIST + b];
    unsigned int k = (P < (unsigned long long)TOPK) ? (unsigned int)P : TOPK;
    ws[H_P] = (unsigned int)P;
    ws[H_K] = k;
    unsigned int thr = 0xFFFFFFFFu;  // collect nothing if no positives
    if (k > 0) {
        unsigned long long cum = 0;
        int b = NBINS - 1;
        for (; b >= 1; --b) { cum += ws[OFF_HIST + b]; if (cum >= k) break; }
        thr = ((unsigned int)b) << 18;   // bucket lower bound (monotone float bits)
    }
    ws[H_THR] = thr;
}

// -------------------- exact top-k over candidates (MSB radix select) ----------
__global__ __launch_bounds__(1024) void select_topk_kernel(unsigned int* __restrict__ ws,
                                                           const float* __restrict__ cand,
                                                           int candCap,
                                                           float* __restrict__ out) {
    __shared__ unsigned int h[256];
    __shared__ float red[1024];
    __shared__ unsigned int sPrefix, sKRem;

    const int tid = threadIdx.x;
    unsigned int C = ws[H_CNT];
    if (C > (unsigned int)candCap) C = (unsigned int)candCap;
    unsigned int k = ws[H_K];
    if (k > C) k = C;
    const float allLoss = ((const float*)ws)[H_SUM] /
                          ((float)NROWS * (float)(NROWS - 1));

    if (k == 0) {  // no positive loss -> fall back to the single largest loss
        if (tid == 0) {
            out[0] = __uint_as_float(ws[H_MAX]);
            out[1] = allLoss;
        }
        return;
    }

    unsigned int prefix = 0, kRem = k;
    for (int rI = 0; rI < 4; ++rI) {
        const unsigned int sh = 24u - 8u * rI;
        const unsigned int known = (rI == 0) ? 0u : (0xFFFFFFFFu << (sh + 8u));
        for (int i = tid; i < 256; i += 1024) h[i] = 0u;
        __syncthreads();
        for (unsigned int i = tid; i < C; i += 1024) {
            unsigned int bits = __float_as_uint(cand[i]);
            if ((bits & known) == prefix) atomicAdd(&h[(bits >> sh) & 0xFFu], 1u);
        }
        __syncthreads();
        if (tid == 0) {
            unsigned int cum = 0, chosen = 0;
            for (int b = 255; b >= 0; --b) {
                unsigned int c = h[b];
                if (cum + c >= kRem) { chosen = (unsigned int)b; break; }
                cum += c;
            }
            sPrefix = prefix | (chosen << sh);
            sKRem = kRem - cum;
        }
        __syncthreads();
        prefix = sPrefix;
        kRem = sKRem;
        __syncthreads();
    }

    // sum of values strictly greater than the kth, plus kRem copies of the kth
    float part = 0.f;
    for (unsigned int i = tid; i < C; i += 1024) {
        if (__float_as_uint(cand[i]) > prefix) part += cand[i];
    }
    red[tid] = part;
    __syncthreads();
    for (int s = 512; s; s >>= 1) {
        if (tid < s) red[tid] += red[tid + s];
        __syncthreads();
    }
    if (tid == 0) {
        float total = red[0] + (float)kRem * __uint_as_float(prefix);
        out[0] = total / (float)k;
        out[1] = allLoss;
    }
}

// -------------------- launch --------------------
extern "C" void kernel_launch(void* const* d_in, const int* in_sizes, int n_in,
                              void* d_out, int out_size, void* d_ws, size_t ws_size,
                              hipStream_t stream) {
    const float* p1 = (const float*)d_in[0];
    const float* p2 = (const float*)d_in[1];
    float* out = (float*)d_out;
    unsigned int* ws = (unsigned int*)d_ws;
    float* cand = (float*)(ws + OFF_CAND);

    long avail = (long)(ws_size / 4) - (long)OFF_CAND;
    int candCap = (avail < 1) ? 1 : ((avail > 65536) ? 65536 : (int)avail);

    init_kernel<<<33, 256, 0, stream>>>(ws);
    prep_kernel<<<NROWS / 8, 256, 0, stream>>>(p1, p2, ws);

    dim3 grid(NROWS / 128, NROWS / 128);
    gemm_kernel<0><<<grid, 256, 0, stream>>>(p1, p2, ws, cand, candCap);
    threshold_kernel<<<1, 64, 0, stream>>>(ws);
    gemm_kernel<1><<<grid, 256, 0, stream>>>(p1, p2, ws, cand, candCap);
    select_topk_kernel<<<1, 1024, 0, stream>>>(ws, cand, candCap, out);
}